// GD_Block_81561428951752
// MI455X (gfx1250) — compile-verified
//
#include <hip/hip_runtime.h>

#define N_NODES 100000
#define N_EDGES 1600000
#define D 64
#define N_TILES (N_NODES / 16)   // 6250, exact

typedef __attribute__((ext_vector_type(2))) float v2f;
typedef __attribute__((ext_vector_type(8))) float v8f;

// ---------------------------------------------------------------------------
// Kernel 1: zero the ax accumulator region (float4 stores, grid-stride free)
// ---------------------------------------------------------------------------
__global__ void zero_ws(float* __restrict__ p, int n4) {
  int i = blockIdx.x * blockDim.x + threadIdx.x;
  if (i < n4) {
    float4 z = {0.f, 0.f, 0.f, 0.f};
    ((float4*)p)[i] = z;
  }
}

// ---------------------------------------------------------------------------
// Kernel 2: ax[dst] += x[src]  (16 threads per edge, 4 floats each)
// ---------------------------------------------------------------------------
__global__ void scatter_add_x(const float* __restrict__ x,
                              const long long* __restrict__ ei,
                              float* __restrict__ ax) {
  int gid = blockIdx.x * blockDim.x + threadIdx.x;
  int e = gid >> 4;
  int l = gid & 15;
  if (e >= N_EDGES) return;
  int src = (int)ei[e];
  int dst = (int)ei[N_EDGES + e];
  float4 xv = *(const float4*)(x + (size_t)src * D + l * 4);
  float* a = ax + (size_t)dst * D + l * 4;
  atomicAdd(a + 0, xv.x);
  atomicAdd(a + 1, xv.y);
  atomicAdd(a + 2, xv.z);
  atomicAdd(a + 3, xv.w);
}

// ---------------------------------------------------------------------------
// Kernel 3: fused WMMA GEMMs.
//   gcn = X@W0 + AX@W1 ; out = gcn/n + ((n-1)/n)*X
//   q = X@Wq ; k = X@Wk ; v = X@Wv      (to workspace)
// One wave per 16-row tile. fp32 WMMA 16x16x4, K accumulated in 16 steps.
// ---------------------------------------------------------------------------
__device__ __forceinline__ v8f wmma_f32(v2f a, v2f b, v8f c) {
  // 8 args: (neg_a, A, neg_b, B, c_mod, C, reuse_a, reuse_b)
  return __builtin_amdgcn_wmma_f32_16x16x4_f32(false, a, false, b, (short)0, c,
                                               false, false);
}

__global__ void fused_gemm(const float* __restrict__ x,
                           const float* __restrict__ ax,
                           const float* __restrict__ W0,
                           const float* __restrict__ W1,
                           const float* __restrict__ Wq,
                           const float* __restrict__ Wk,
                           const float* __restrict__ Wv,
                           float* __restrict__ out,
                           float* __restrict__ q,
                           float* __restrict__ k,
                           float* __restrict__ v) {
  const int wave = (blockIdx.x * blockDim.x + threadIdx.x) >> 5;
  if (wave >= N_TILES) return;          // wave-uniform: EXEC stays all-ones
  const int lane = threadIdx.x & 31;
  const int col  = lane & 15;
  const int half = lane >> 4;
  const int row0 = wave * 16;
  const int arow = row0 + col;          // A-matrix row owned by this lane

  const float inv_n = 1.0f / (float)N_NODES;
  const float selfw = (float)(N_NODES - 1) * inv_n;

  for (int n0 = 0; n0 < D; n0 += 16) {
    v8f cg = {}; v8f cq = {}; v8f ck = {}; v8f cv = {};
#pragma unroll
    for (int k0 = 0; k0 < D; k0 += 4) {
      const int kk = k0 + 2 * half;     // lane-half selects K pair (ISA layout)
      // A fragments: 2 consecutive K values of this lane's row (contiguous)
      v2f a_x  = *(const v2f*)(x  + (size_t)arow * D + kk);
      v2f a_ax = *(const v2f*)(ax + (size_t)arow * D + kk);
      // B fragments: rows kk, kk+1 of W, column n0+col
      v2f b0 = { W0[kk * D + n0 + col], W0[(kk + 1) * D + n0 + col] };
      v2f b1 = { W1[kk * D + n0 + col], W1[(kk + 1) * D + n0 + col] };
      v2f bq = { Wq[kk * D + n0 + col], Wq[(kk + 1) * D + n0 + col] };
      v2f bk = { Wk[kk * D + n0 + col], Wk[(kk + 1) * D + n0 + col] };
      v2f bv = { Wv[kk * D + n0 + col], Wv[(kk + 1) * D + n0 + col] };
      cg = wmma_f32(a_x,  b0, cg);
      cg = wmma_f32(a_ax, b1, cg);
      cq = wmma_f32(a_x,  bq, cq);
      ck = wmma_f32(a_x,  bk, ck);
      cv = wmma_f32(a_x,  bv, cv);
    }
    // C/D layout: VGPR r -> M = r + 8*half, N = col
#pragma unroll
    for (int r = 0; r < 8; ++r) {
      const int m = row0 + r + 8 * half;
      const size_t idx = (size_t)m * D + n0 + col;
      out[idx] = cg[r] * inv_n + selfw * x[idx];
      q[idx] = cq[r];
      k[idx] = ck[r];
      v[idx] = cv[r];
    }
  }
}

// ---------------------------------------------------------------------------
// Kernel 4: per-edge attention (one wave32 per edge)
//   score = <q[dst], k[src]> ; out[dst] -= score/n^3 * v[src]
// ---------------------------------------------------------------------------
__global__ void attn_edges(const float* __restrict__ q,
                           const float* __restrict__ k,
                           const float* __restrict__ v,
                           const long long* __restrict__ ei2,
                           float* __restrict__ out) {
  int gid = blockIdx.x * blockDim.x + threadIdx.x;
  int e = gid >> 5;
  int lane = gid & 31;                  // blockDim is a multiple of 32
  if (e >= N_EDGES) return;             // wave-uniform
  int s = (int)ei2[e];
  int d = (int)ei2[N_EDGES + e];
  const float* qd = q + (size_t)d * D;
  const float* ks = k + (size_t)s * D;
  float p = qd[lane] * ks[lane] + qd[32 + lane] * ks[32 + lane];
#pragma unroll
  for (int off = 16; off > 0; off >>= 1) p += __shfl_xor(p, off, 32);
  const float nn = (float)N_NODES;
  const float sc = p * (-1.0f / (nn * nn * nn));
  const float* vs = v + (size_t)s * D;
  atomicAdd(&out[(size_t)d * D + lane],      sc * vs[lane]);
  atomicAdd(&out[(size_t)d * D + 32 + lane], sc * vs[32 + lane]);
}

// ---------------------------------------------------------------------------
extern "C" void kernel_launch(void* const* d_in, const int* in_sizes, int n_in,
                              void* d_out, int out_size, void* d_ws, size_t ws_size,
                              hipStream_t stream) {
  const float*     x   = (const float*)d_in[0];
  const long long* ei  = (const long long*)d_in[1];
  const long long* ei2 = (const long long*)d_in[2];
  const float*     W0  = (const float*)d_in[3];
  const float*     W1  = (const float*)d_in[4];
  const float*     Wq  = (const float*)d_in[5];
  const float*     Wk  = (const float*)d_in[6];
  const float*     Wv  = (const float*)d_in[7];
  float* out = (float*)d_out;

  float* ws = (float*)d_ws;
  const size_t ND = (size_t)N_NODES * D;
  float* ax = ws;
  float* qb = ws + ND;
  float* kb = ws + 2 * ND;
  float* vb = ws + 3 * ND;

  // 1) zero ax accumulator
  {
    int n4 = (int)(ND / 4);
    zero_ws<<<(n4 + 255) / 256, 256, 0, stream>>>(ax, n4);
  }
  // 2) ax = segment_sum(x[src], dst)
  {
    long long work = (long long)N_EDGES * 16;
    int blocks = (int)((work + 255) / 256);
    scatter_add_x<<<blocks, 256, 0, stream>>>(x, ei, ax);
  }
  // 3) fused WMMA GEMMs -> out (= x_1), q/k/v workspace
  {
    int waves = N_TILES;
    int blocks = (waves * 32 + 255) / 256;
    fused_gemm<<<blocks, 256, 0, stream>>>(x, ax, W0, W1, Wq, Wk, Wv,
                                           out, qb, kb, vb);
  }
  // 4) out -= score/n^3 * v[src]  (per edge)
  {
    long long work = (long long)N_EDGES * 32;
    int blocks = (int)((work + 255) / 256);
    attn_edges<<<blocks, 256, 0, stream>>>(qb, kb, vb, ei2, out);
  }
}